// comb_filter_15298673508463
// MI455X (gfx1250) — compile-verified
//
#include <hip/hip_runtime.h>

#ifndef __has_builtin
#define __has_builtin(x) 0
#endif

#define NSAMP   192000
#define BATCH   2
#define TFR     497
#define PADL    768
#define FRAMEL  3072
#define HOP     384
#define NFFT    1536
#define KLEN    1537
#define NPITCH  226
#define MAXT    8

typedef int v4i __attribute__((vector_size(16)));
typedef __attribute__((address_space(1))) v4i* g_v4i_p;   // global int4*
typedef __attribute__((address_space(3))) v4i* l_v4i_p;   // LDS int4*
typedef __attribute__((address_space(3))) void LVoid;

// ---------------- async global -> LDS (CDNA5) ----------------
__device__ __forceinline__ void async_load_b128(void* lds_dst, const float* gsrc) {
#if __has_builtin(__builtin_amdgcn_global_load_async_to_lds_b128)
    __builtin_amdgcn_global_load_async_to_lds_b128(
        (g_v4i_p)gsrc, (l_v4i_p)lds_dst, /*offset=*/0, /*cpol=*/0);
#else
    unsigned loff = (unsigned)(unsigned long long)(LVoid*)lds_dst;
    asm volatile("global_load_async_to_lds_b128 %0, %1, off"
                 :: "v"(loff), "v"(gsrc) : "memory");
#endif
}

__device__ __forceinline__ void wait_async0() {
#if __has_builtin(__builtin_amdgcn_s_wait_asynccnt)
    __builtin_amdgcn_s_wait_asynccnt(0);
#else
    asm volatile("s_wait_asynccnt 0" ::: "memory");
#endif
}

__device__ __forceinline__ unsigned ballot32(bool p) {
#if __has_builtin(__builtin_amdgcn_ballot_w32)
    return __builtin_amdgcn_ballot_w32(p);
#else
    return (unsigned)__ballot(p);
#endif
}

// Deterministic wave32 compaction of non-zero taps of one conv_w row
// (order-preserving in k, so summation order is stable run-to-run).
__device__ __forceinline__ int extract_row(const float* __restrict__ wrow,
                                           int lane, int* tk, float* tw) {
    int cnt = 0;
    for (int base = 0; base < KLEN; base += 32) {
        int k = base + lane;
        float v = (k < KLEN) ? wrow[k] : 0.0f;
        bool nz = (v != 0.0f);
        unsigned mask = ballot32(nz);
        int pos = cnt + __popc(mask & ((1u << lane) - 1u));
        if (nz && pos < MAXT) { tk[pos] = k; tw[pos] = v; }
        cnt += __popc(mask);
    }
    return (cnt > MAXT) ? MAXT : cnt;
}

// ---------------- main kernel: one frame per block ----------------
__global__ void __launch_bounds__(256)
comb_main(const float* __restrict__ x, const int* __restrict__ pitch,
          const float* __restrict__ conv_w,
          const int* __restrict__ tapn, const int* __restrict__ tapk,
          const float* __restrict__ tapw, const float* __restrict__ window,
          int precomp, float* __restrict__ out) {
    __shared__ float s_fr[FRAMEL];
    __shared__ int   s_tk[MAXT];
    __shared__ float s_tw[MAXT];
    __shared__ int   s_nt;

    const int bt  = blockIdx.x;               // b*TFR + t
    const int b   = bt / TFR;
    const int t   = bt % TFR;
    const int tid = threadIdx.x;

    const int   start    = t * HOP - PADL;     // x-index of frame element 0
    const bool  interior = (start >= 0) && (start + FRAMEL <= NSAMP);
    const float* xb      = x + b * NSAMP;

    // Kick off the DMA staging FIRST; tap setup overlaps the async copies.
    if (interior) {
        // Stage 3072 floats = 768 x b128 async copies; 3 per thread.
        #pragma unroll
        for (int i = 0; i < 3; ++i) {
            int q = tid + i * 256;             // float4 index
            async_load_b128(&s_fr[q * 4], xb + start + q * 4);
        }
    }

    const int p = pitch[bt];
    if (precomp) {
        if (tid == 0)   s_nt = tapn[p];
        if (tid < MAXT) { s_tk[tid] = tapk[p * MAXT + tid];
                          s_tw[tid] = tapw[p * MAXT + tid]; }
    } else if (tid < 32) {
        int n = extract_row(conv_w + p * KLEN, tid, s_tk, s_tw);
        if (tid == 0) s_nt = n;
    }

    if (interior) wait_async0();
    __syncthreads();

    const int obase = bt * NFFT;
    const int nt    = s_nt;

    if (interior) {
        #pragma unroll
        for (int i = 0; i < NFFT / 256; ++i) {
            int l = tid + i * 256;
            float win = precomp ? window[l]
                                : (0.5f - 0.5f * cosf(6.2831855f * (float)l *
                                                      (1.0f / (float)NFFT)));
            float acc = 0.0f;
            for (int j = 0; j < nt; ++j)
                acc += s_tw[j] * s_fr[l + s_tk[j]];
            out[obase + l] = acc * win;
        }
    } else {
        #pragma unroll
        for (int i = 0; i < NFFT / 256; ++i) {
            int l = tid + i * 256;
            float win = precomp ? window[l]
                                : (0.5f - 0.5f * cosf(6.2831855f * (float)l *
                                                      (1.0f / (float)NFFT)));
            float acc = 0.0f;
            for (int j = 0; j < nt; ++j) {
                int idx = start + l + s_tk[j];
                float v = (idx >= 0 && idx < NSAMP) ? xb[idx] : 0.0f;
                acc += s_tw[j] * v;
            }
            out[obase + l] = acc * win;
        }
    }
}

// ---------------- kernel 1: tap extraction + window table ----------------
__global__ void __launch_bounds__(32)
extract_kernel(const float* __restrict__ conv_w, int* __restrict__ tapn,
               int* __restrict__ tapk, float* __restrict__ tapw,
               float* __restrict__ window) {
    const int row  = blockIdx.x;
    const int lane = threadIdx.x;
    if (row < NPITCH) {
        int n = extract_row(conv_w + row * KLEN, lane,
                            tapk + row * MAXT, tapw + row * MAXT);
        if (lane == 0) tapn[row] = n;
    }
    int gid = row * 32 + lane;
    if (gid < NFFT) {
        float ph = 6.2831855f * (float)gid * (1.0f / (float)NFFT);
        window[gid] = 0.5f - 0.5f * cosf(ph);
    }
}

// ---------------- host launcher ----------------
extern "C" void kernel_launch(void* const* d_in, const int* in_sizes, int n_in,
                              void* d_out, int out_size, void* d_ws, size_t ws_size,
                              hipStream_t stream) {
    const float* x      = (const float*)d_in[0];
    const int*   pitch  = (const int*)d_in[1];
    const float* conv_w = (const float*)d_in[2];
    float*       out    = (float*)d_out;

    const size_t need = sizeof(float) * NFFT
                      + sizeof(int)   * NPITCH
                      + sizeof(int)   * NPITCH * MAXT
                      + sizeof(float) * NPITCH * MAXT;
    const int precomp = (ws_size >= need) ? 1 : 0;

    float* w_window = (float*)d_ws;
    int*   w_tapn   = (int*)(w_window + NFFT);
    int*   w_tapk   = w_tapn + NPITCH;
    float* w_tapw   = (float*)(w_tapk + NPITCH * MAXT);

    if (precomp) {
        extract_kernel<<<NPITCH, 32, 0, stream>>>(conv_w, w_tapn, w_tapk,
                                                  w_tapw, w_window);
    }
    comb_main<<<BATCH * TFR, 256, 0, stream>>>(x, pitch, conv_w, w_tapn, w_tapk,
                                               w_tapw, w_window, precomp, out);
}